// MultiHopGAT_67559835566303
// MI455X (gfx1250) — compile-verified
//
#include <hip/hip_runtime.h>

// ---------------- problem constants (fixed by reference) ----------------
#define NN      50000
#define F_IN    256
#define NH      8
#define HPC     32
#define HIDDEN  256      // NH*HPC
#define NOUT    64
#define NE      800000
#define NEG_SLOPE 0.2f

// ---------------- workspace layout (float offsets) ----------------
static const size_t OFF_H   = 0;                      // 50000*256  (h per hop; h2 later)
static const size_t OFF_ACC = 12800000;               // 50000*256  (aggregation out; acc2 later)
static const size_t OFF_S   = 25600000;               // 50000*256  (skip accum -> "first")
static const size_t OFF_FIN = 38400000;               // 50000*64   (first@final_W + final_b)
static const size_t OFF_AS  = 41600000;               // 50000*8
static const size_t OFF_AD  = 42000000;               // 50000*8
static const size_t OFF_M   = 42400000;               // 50000*8
static const size_t OFF_DEN = 42800000;               // 50000*8
// total 43.2M floats = 172.8 MB

typedef float v2f __attribute__((ext_vector_type(2)));
typedef float v8f __attribute__((ext_vector_type(8)));

// ---------------- fp32 WMMA GEMM: C[M,N] = A[M,K] @ B[K,N] (+bias) ----------------
// One wave per 16(M) x 64(N) strip: 4 accumulators, A fragment reused across
// 4 V_WMMA_F32_16X16X4_F32 issues per K-step (2.5 loaded dwords per WMMA).
// A 16x4 layout: lane holds row M=lane&15, K-pair base = (lane>>4)*2 (ISA 7.12.2).
// B 4x16 layout: lane holds col N=lane&15, K-pair base = (lane>>4)*2.
// C/D: lane&15 = N, VGPR v -> M = v + 8*(lane>>4).
// Requires N % 64 == 0 (N in {64, 256} here), K % 4 == 0, M % 16 == 0.
__global__ __launch_bounds__(32) void wmma_gemm_f32(
    const float* __restrict__ A, const float* __restrict__ B,
    const float* __restrict__ bias, float* __restrict__ C,
    int M, int N, int K)
{
    const int lane = threadIdx.x;           // 0..31, full wave (EXEC all ones)
    const int m0   = blockIdx.x * 16;
    const int n0   = blockIdx.y * 64;
    const int r    = lane & 15;
    const int koff = (lane >> 4) * 2;

    const float* Arow = A + (size_t)(m0 + r) * K;
    const float* B0   = B + n0 + r;

    v8f acc0 = {}, acc1 = {}, acc2 = {}, acc3 = {};
    for (int k0 = 0; k0 < K; k0 += 4) {
        // prefetch A one chunk ahead (global_prefetch_b8)
        if ((k0 & 63) == 0 && k0 + 64 < K)
            __builtin_prefetch(Arow + k0 + 64, 0, 1);

        v2f a;
        a.x = Arow[k0 + koff];
        a.y = Arow[k0 + koff + 1];

        const float* Brow0 = B0 + (size_t)(k0 + koff) * N;
        const float* Brow1 = B0 + (size_t)(k0 + koff + 1) * N;
        v2f b0, b1, b2, b3;
        b0.x = Brow0[0];  b0.y = Brow1[0];
        b1.x = Brow0[16]; b1.y = Brow1[16];
        b2.x = Brow0[32]; b2.y = Brow1[32];
        b3.x = Brow0[48]; b3.y = Brow1[48];

        acc0 = __builtin_amdgcn_wmma_f32_16x16x4_f32(false, a, false, b0, (short)0, acc0, false, false);
        acc1 = __builtin_amdgcn_wmma_f32_16x16x4_f32(false, a, false, b1, (short)0, acc1, false, false);
        acc2 = __builtin_amdgcn_wmma_f32_16x16x4_f32(false, a, false, b2, (short)0, acc2, false, false);
        acc3 = __builtin_amdgcn_wmma_f32_16x16x4_f32(false, a, false, b3, (short)0, acc3, false, false);
    }

    const int col   = lane & 15;
    const int rbase = (lane >> 4) * 8;
#define STORE_TILE(ACCV, T)                                                     \
    {                                                                           \
        const float bv = bias ? bias[n0 + 16 * (T) + col] : 0.0f;               \
        for (int v = 0; v < 8; ++v)                                             \
            C[(size_t)(m0 + rbase + v) * N + n0 + 16 * (T) + col] = ACCV[v] + bv; \
    }
    STORE_TILE(acc0, 0)
    STORE_TILE(acc1, 1)
    STORE_TILE(acc2, 2)
    STORE_TILE(acc3, 3)
#undef STORE_TILE
}

// ---------------- helpers ----------------
__device__ inline void edge_sd(const int* __restrict__ ei, int e, int E, int& s, int& d)
{
    if (e < E) { s = ei[e]; d = ei[E + e]; }
    else       { s = e - E; d = e - E; }     // appended self loops
}

__device__ inline float lrelu(float v) { return v > 0.0f ? v : NEG_SLOPE * v; }

__device__ inline void atomicMaxFloat(float* addr, float v)
{
    if (v >= 0.0f) atomicMax((int*)addr, __float_as_int(v));
    else           atomicMin((unsigned int*)addr, __float_as_uint(v));
}

__global__ void fill_kernel(float* __restrict__ p, float v, int n)
{
    int i = blockIdx.x * blockDim.x + threadIdx.x;
    if (i < n) p[i] = v;
}

// a_s[n,h] = <h[n,h,:], att_src[h,:]>, same for a_d
__global__ void attn_kernel(const float* __restrict__ h,
                            const float* __restrict__ att_src,
                            const float* __restrict__ att_dst,
                            float* __restrict__ a_s, float* __restrict__ a_d,
                            int H, int C)
{
    int idx = blockIdx.x * blockDim.x + threadIdx.x;   // node*H + head
    if (idx >= NN * H) return;
    int node = idx / H, head = idx % H;
    const float* hp = h + (size_t)node * H * C + (size_t)head * C;
    const float* as = att_src + head * C;
    const float* ad = att_dst + head * C;
    float ss = 0.0f, sd = 0.0f;
    for (int c = 0; c < C; ++c) { float v = hp[c]; ss += v * as[c]; sd += v * ad[c]; }
    a_s[idx] = ss; a_d[idx] = sd;
}

__global__ void edge_max_kernel(const int* __restrict__ ei,
                                const float* __restrict__ a_s,
                                const float* __restrict__ a_d,
                                float* __restrict__ m, int H)
{
    int idx = blockIdx.x * blockDim.x + threadIdx.x;
    if (idx >= (NE + NN) * H) return;
    int e = idx / H, h = idx % H;
    int s, d; edge_sd(ei, e, NE, s, d);
    float v = lrelu(a_s[s * H + h] + a_d[d * H + h]);
    atomicMaxFloat(&m[d * H + h], v);
}

__global__ void edge_sum_kernel(const int* __restrict__ ei,
                                const float* __restrict__ a_s,
                                const float* __restrict__ a_d,
                                const float* __restrict__ m,
                                float* __restrict__ den, int H)
{
    int idx = blockIdx.x * blockDim.x + threadIdx.x;
    if (idx >= (NE + NN) * H) return;
    int e = idx / H, h = idx % H;
    int s, d; edge_sd(ei, e, NE, s, d);
    float v = lrelu(a_s[s * H + h] + a_d[d * H + h]);
    atomicAdd(&den[d * H + h], expf(v - m[d * H + h]));
}

// one wave per edge: acc[dst, :] += h[src, :] * alpha[head(c)]
__global__ __launch_bounds__(256) void edge_agg_kernel(
    const int* __restrict__ ei, const float* __restrict__ h,
    const float* __restrict__ a_s, const float* __restrict__ a_d,
    const float* __restrict__ m, const float* __restrict__ den,
    float* __restrict__ acc, int H, int C, int F)
{
    int wave = blockIdx.x * (blockDim.x >> 5) + (threadIdx.x >> 5);
    int lane = threadIdx.x & 31;
    if (wave >= NE + NN) return;
    int s, d; edge_sd(ei, wave, NE, s, d);
    const float* hs = h + (size_t)s * F;
    float* ap = acc + (size_t)d * F;
    // pull the (L2-resident) source feature row toward the WGP while the
    // attention coefficients are computed (global_prefetch_b8, per-lane addr)
    __builtin_prefetch(hs + lane * (F / 32), 0, 0);
    for (int j = 0; j * 32 < F; ++j) {
        int c = j * 32 + lane;
        int head = c / C;       // C=32,H=8 -> head=j ; C=64,H=1 -> head=0
        float e = lrelu(a_s[s * H + head] + a_d[d * H + head]);
        float alpha = expf(e - m[d * H + head]) / (den[d * H + head] + 1e-16f);
        atomicAdd(&ap[c], hs[c] * alpha);
    }
}

// S += w[hop] * elu(acc + b1[hop])   (S already holds skip projection)
__global__ void combine_kernel(const float* __restrict__ acc,
                               const float* __restrict__ b1h,
                               const float* __restrict__ hop_att,
                               float* __restrict__ S, int hop)
{
    int i = blockIdx.x * blockDim.x + threadIdx.x;
    if (i >= NN * HIDDEN) return;
    float ha0 = hop_att[0], ha1 = hop_att[1];
    float mx = fmaxf(ha0, ha1);
    float e0 = expf(ha0 - mx), e1 = expf(ha1 - mx);
    float w = (hop == 0 ? e0 : e1) / (e0 + e1);
    float v = acc[i] + b1h[i % HIDDEN];
    v = v > 0.0f ? v : (expf(v) - 1.0f);       // ELU
    S[i] += w * v;
}

// fin[i] += acc2[i] + b2[c]   (fin already holds first@final_W + final_b)
__global__ void add3_kernel(const float* __restrict__ acc2,
                            const float* __restrict__ b2,
                            float* __restrict__ fin)
{
    int i = blockIdx.x * blockDim.x + threadIdx.x;
    if (i >= NN * NOUT) return;
    fin[i] += acc2[i] + b2[i % NOUT];
}

// wave-per-row LayerNorm over width W (wave32 shuffle reduction)
template <int W>
__global__ __launch_bounds__(256) void layernorm_kernel(
    const float* __restrict__ in, const float* __restrict__ g,
    const float* __restrict__ b, float* __restrict__ out)
{
    int wave = blockIdx.x * (blockDim.x >> 5) + (threadIdx.x >> 5);
    int lane = threadIdx.x & 31;
    if (wave >= NN) return;
    const float* row = in + (size_t)wave * W;
    constexpr int PL = W / 32;
    float vals[PL];
    float s = 0.0f;
    for (int j = 0; j < PL; ++j) { vals[j] = row[lane + 32 * j]; s += vals[j]; }
    for (int off = 16; off; off >>= 1) s += __shfl_xor(s, off, 32);
    float mu = s / (float)W;
    float vs = 0.0f;
    for (int j = 0; j < PL; ++j) { float d = vals[j] - mu; vs += d * d; }
    for (int off = 16; off; off >>= 1) vs += __shfl_xor(vs, off, 32);
    float inv = rsqrtf(vs / (float)W + 1e-5f);
    for (int j = 0; j < PL; ++j) {
        int c = lane + 32 * j;
        out[(size_t)wave * W + c] = (vals[j] - mu) * inv * g[c] + b[c];
    }
}

__global__ void hopw_kernel(const float* __restrict__ ha, float* __restrict__ out)
{
    if (threadIdx.x == 0) {
        float mx = fmaxf(ha[0], ha[1]);
        float e0 = expf(ha[0] - mx), e1 = expf(ha[1] - mx);
        out[0] = e0 / (e0 + e1);
        out[1] = e1 / (e0 + e1);
    }
}

// ---------------- driver ----------------
extern "C" void kernel_launch(void* const* d_in, const int* in_sizes, int n_in,
                              void* d_out, int out_size, void* d_ws, size_t ws_size,
                              hipStream_t stream)
{
    const float* x      = (const float*)d_in[0];
    const int*   ei1    = (const int*)  d_in[1];
    const int*   ei2    = (const int*)  d_in[2];
    const float* W1     = (const float*)d_in[3];
    const float* as1    = (const float*)d_in[4];
    const float* ad1    = (const float*)d_in[5];
    const float* b1     = (const float*)d_in[6];
    const float* W2     = (const float*)d_in[7];
    const float* as2    = (const float*)d_in[8];
    const float* ad2    = (const float*)d_in[9];
    const float* b2     = (const float*)d_in[10];
    const float* hopatt = (const float*)d_in[11];
    const float* skipW  = (const float*)d_in[12];
    const float* skipb  = (const float*)d_in[13];
    const float* finW   = (const float*)d_in[14];
    const float* finb   = (const float*)d_in[15];
    const float* ln1g   = (const float*)d_in[16];
    const float* ln1b   = (const float*)d_in[17];
    const float* ln2g   = (const float*)d_in[18];
    const float* ln2b   = (const float*)d_in[19];

    float* out = (float*)d_out;
    float* ws  = (float*)d_ws;

    float* Hb  = ws + OFF_H;
    float* ACC = ws + OFF_ACC;
    float* S   = ws + OFF_S;
    float* FIN = ws + OFF_FIN;
    float* AS  = ws + OFF_AS;
    float* AD  = ws + OFF_AD;
    float* MM  = ws + OFF_M;
    float* DEN = ws + OFF_DEN;

    const int TB = 256;
    const int Etot = NE + NN;
    auto cdiv = [](int a, int b) { return (a + b - 1) / b; };

    // 1) skip projection: S = x @ skip_W + skip_b
    wmma_gemm_f32<<<dim3(NN / 16, HIDDEN / 64), 32, 0, stream>>>(
        x, skipW, skipb, S, NN, HIDDEN, F_IN);

    // 2) per-hop GAT (heads=8, concat)
    const int* eis[2] = { ei1, ei2 };
    for (int hop = 0; hop < 2; ++hop) {
        const int* ei = eis[hop];
        // h = x @ W1[hop]   (bias applied post-aggregation per reference)
        wmma_gemm_f32<<<dim3(NN / 16, HIDDEN / 64), 32, 0, stream>>>(
            x, W1 + (size_t)hop * F_IN * HIDDEN, nullptr, Hb, NN, HIDDEN, F_IN);
        // attention coefficients
        attn_kernel<<<cdiv(NN * NH, TB), TB, 0, stream>>>(
            Hb, as1 + hop * NH * HPC, ad1 + hop * NH * HPC, AS, AD, NH, HPC);
        // init segment state + accumulator
        fill_kernel<<<cdiv(NN * NH, TB), TB, 0, stream>>>(MM, -__builtin_huge_valf(), NN * NH);
        fill_kernel<<<cdiv(NN * NH, TB), TB, 0, stream>>>(DEN, 0.0f, NN * NH);
        fill_kernel<<<cdiv(NN * HIDDEN, TB), TB, 0, stream>>>(ACC, 0.0f, NN * HIDDEN);
        // segment softmax + aggregation
        edge_max_kernel<<<cdiv(Etot * NH, TB), TB, 0, stream>>>(ei, AS, AD, MM, NH);
        edge_sum_kernel<<<cdiv(Etot * NH, TB), TB, 0, stream>>>(ei, AS, AD, MM, DEN, NH);
        edge_agg_kernel<<<cdiv(Etot, TB / 32), TB, 0, stream>>>(
            ei, Hb, AS, AD, MM, DEN, ACC, NH, HPC, HIDDEN);
        // S += w[hop] * elu(ACC + b1[hop])
        combine_kernel<<<cdiv(NN * HIDDEN, TB), TB, 0, stream>>>(
            ACC, b1 + hop * HIDDEN, hopatt, S, hop);
    }

    // 3) LayerNorm1 in place -> S becomes "first"
    layernorm_kernel<HIDDEN><<<cdiv(NN, TB / 32), TB, 0, stream>>>(S, ln1g, ln1b, S);

    // 4) conv2 (heads=1, concat=False): h2 = first @ W2
    wmma_gemm_f32<<<dim3(NN / 16, NOUT / 64), 32, 0, stream>>>(
        S, W2, nullptr, Hb, NN, NOUT, F_IN);
    attn_kernel<<<cdiv(NN, TB), TB, 0, stream>>>(Hb, as2, ad2, AS, AD, 1, NOUT);
    fill_kernel<<<cdiv(NN, TB), TB, 0, stream>>>(MM, -__builtin_huge_valf(), NN);
    fill_kernel<<<cdiv(NN, TB), TB, 0, stream>>>(DEN, 0.0f, NN);
    fill_kernel<<<cdiv(NN * NOUT, TB), TB, 0, stream>>>(ACC, 0.0f, NN * NOUT);
    edge_max_kernel<<<cdiv(Etot, TB), TB, 0, stream>>>(ei1, AS, AD, MM, 1);
    edge_sum_kernel<<<cdiv(Etot, TB), TB, 0, stream>>>(ei1, AS, AD, MM, DEN, 1);
    edge_agg_kernel<<<cdiv(Etot, TB / 32), TB, 0, stream>>>(
        ei1, Hb, AS, AD, MM, DEN, ACC, 1, NOUT, NOUT);

    // 5) final residual: FIN = first @ final_W + final_b ; FIN += ACC + b2
    wmma_gemm_f32<<<dim3(NN / 16, NOUT / 64), 32, 0, stream>>>(
        S, finW, finb, FIN, NN, NOUT, F_IN);
    add3_kernel<<<cdiv(NN * NOUT, TB), TB, 0, stream>>>(ACC, b2, FIN);

    // 6) LayerNorm2 -> d_out[0 : NN*NOUT]
    layernorm_kernel<NOUT><<<cdiv(NN, TB / 32), TB, 0, stream>>>(FIN, ln2g, ln2b, out);

    // 7) hop weights -> d_out tail
    hopw_kernel<<<1, 32, 0, stream>>>(hopatt, out + (size_t)NN * NOUT);
}